// HCFCGNN_34763465294566
// MI455X (gfx1250) — compile-verified
//
#include <hip/hip_runtime.h>
#include <hip/hip_bf16.h>

typedef __attribute__((ext_vector_type(2))) float v2f;
typedef __attribute__((ext_vector_type(8))) float v8f;

#define DIN 12
#define HID 128
#define CP 16   // padded class dim (C=13 -> 16)

// ---------------- degree / norm ----------------

__global__ void count_src_kernel(const int* __restrict__ src, unsigned* __restrict__ cnt, int E) {
    int e = blockIdx.x * blockDim.x + threadIdx.x;
    if (e < E) atomicAdd(&cnt[src[e]], 1u);
}

__global__ void make_dinv_kernel(float* __restrict__ dinv, int N) {
    int i = blockIdx.x * blockDim.x + threadIdx.x;
    if (i < N) {
        unsigned c = ((const unsigned*)dinv)[i];
        dinv[i] = rsqrtf((float)(c + 1u));   // +1 self loop
    }
}

__global__ void make_norm_kernel(const int* __restrict__ src, const int* __restrict__ dst,
                                 const float* __restrict__ dinv, float* __restrict__ norm, int E) {
    int e = blockIdx.x * blockDim.x + threadIdx.x;
    if (e < E) norm[e] = dinv[src[e]] * dinv[dst[e]];
}

// ---------------- GEMM1: t1 = x @ W1^T + b1   (N x 12) * (12 x 128) ----------------
// block = 256 threads = 8 waves; block covers 16 nodes, wave w covers out cols [16w,16w+16)

__global__ void gemm1_wmma(const float* __restrict__ x, const float* __restrict__ W1,
                           const float* __restrict__ b1, float* __restrict__ out, int N) {
    const int wave = threadIdx.x >> 5;
    const int lane = threadIdx.x & 31;
    const int half = lane >> 4;
    const int l    = lane & 15;
    const int nb   = blockIdx.x * 16;
    const int ob   = wave * 16;

    const int arow = min(nb + l, N - 1);      // clamp: keep EXEC full for WMMA
    v8f acc = {};
#pragma unroll
    for (int kk = 0; kk < DIN; kk += 4) {
        const int k0 = kk + 2 * half;
        v2f a, b;
        const float* xp = x + (size_t)arow * DIN + k0;
        a.x = xp[0]; a.y = xp[1];
        const float* wp = W1 + (size_t)(ob + l) * DIN + k0;
        b.x = wp[0]; b.y = wp[1];
        acc = __builtin_amdgcn_wmma_f32_16x16x4_f32(false, a, false, b, (short)0, acc, false, false);
    }
    const float bias = b1[ob + l];
    if (nb + 16 <= N) {                       // uniform fast path: plain coalesced stores
#pragma unroll
        for (int r = 0; r < 8; ++r)
            out[(size_t)(nb + r + 8 * half) * HID + ob + l] = acc[r] + bias;
    } else {
#pragma unroll
        for (int r = 0; r < 8; ++r) {
            const int row = nb + r + 8 * half;
            if (row < N) out[(size_t)row * HID + ob + l] = acc[r] + bias;
        }
    }
}

// ---------------- GEMM2: t2 = relu(g) @ W2^T + b2  (128x128), async-LDS-staged A ----------------

__global__ void gemm2_wmma(const float* __restrict__ g, const float* __restrict__ W2,
                           const float* __restrict__ b2, float* __restrict__ out, int N) {
    __shared__ float As[16 * 132];           // stride 132 -> conflict-free column reads; base offset 0
    const int nb  = blockIdx.x * 16;
    const int tid = threadIdx.x;
    const bool full = (nb + 16 <= N);        // uniform across block

    if (full) {
        // 16x128 floats = 8KB staged via ASYNCcnt-tracked LDS DMA, 2 x b128 per thread
        for (int i = tid * 4; i < 16 * HID; i += 256 * 4) {
            const int r = i >> 7, c = i & 127;
            const unsigned ldsoff = (unsigned)((r * 132 + c) * 4);  // As is at LDS offset 0
            const float* gp = g + (size_t)(nb + r) * HID + c;
            asm volatile("global_load_async_to_lds_b128 %0, %1, off"
                         :: "v"(ldsoff), "v"(gp) : "memory");
        }
        asm volatile("s_wait_asynccnt 0x0" ::: "memory");
    } else {                                  // ragged tail: manual zero-filled staging
        for (int i = tid; i < 16 * HID; i += 256) {
            const int r = i >> 7, c = i & 127;
            float v = 0.0f;
            if (nb + r < N) v = g[(size_t)(nb + r) * HID + c];
            As[r * 132 + c] = v;
        }
    }
    __syncthreads();

    const int wave = tid >> 5;
    const int lane = tid & 31;
    const int half = lane >> 4;
    const int l    = lane & 15;
    const int ob   = wave * 16;

    v8f acc = {};
#pragma unroll 4
    for (int kk = 0; kk < HID; kk += 4) {
        const int k0 = kk + 2 * half;
        v2f a, b;
        a.x = fmaxf(As[l * 132 + k0], 0.0f);      // ReLU fused at LDS read
        a.y = fmaxf(As[l * 132 + k0 + 1], 0.0f);
        const float* wp = W2 + (size_t)(ob + l) * HID + k0;
        b.x = wp[0]; b.y = wp[1];
        acc = __builtin_amdgcn_wmma_f32_16x16x4_f32(false, a, false, b, (short)0, acc, false, false);
    }
    const float bias = b2[ob + l];
    if (full) {
#pragma unroll
        for (int r = 0; r < 8; ++r)
            out[(size_t)(nb + r + 8 * half) * HID + ob + l] = acc[r] + bias;
    } else {
#pragma unroll
        for (int r = 0; r < 8; ++r) {
            const int row = nb + r + 8 * half;
            if (row < N) out[(size_t)row * HID + ob + l] = acc[r] + bias;
        }
    }
}

// ---------------- GEMM3: t3 = relu(g) @ W3^T + b3  -> [N, 16] (cols >= C are zero) ------------

__global__ void gemm3_wmma(const float* __restrict__ g, const float* __restrict__ W3,
                           const float* __restrict__ b3, float* __restrict__ out, int N, int C) {
    const int wave = threadIdx.x >> 5;
    const int lane = threadIdx.x & 31;
    const int nb   = (blockIdx.x * 8 + wave) * 16;
    if (nb >= N) return;                      // uniform per wave: EXEC stays full
    const int half = lane >> 4;
    const int l    = lane & 15;

    const int arow  = min(nb + l, N - 1);
    const bool cval = (l < C);
    v8f acc = {};
#pragma unroll 4
    for (int kk = 0; kk < HID; kk += 4) {
        const int k0 = kk + 2 * half;
        v2f a, b;
        const float* gp = g + (size_t)arow * HID + k0;
        a.x = fmaxf(gp[0], 0.0f);
        a.y = fmaxf(gp[1], 0.0f);
        float bx = 0.0f, by = 0.0f;
        if (cval) {
            const float* wp = W3 + (size_t)l * HID + k0;
            bx = wp[0]; by = wp[1];
        }
        b.x = bx; b.y = by;
        acc = __builtin_amdgcn_wmma_f32_16x16x4_f32(false, a, false, b, (short)0, acc, false, false);
    }
    const float bias = cval ? b3[l] : 0.0f;
    if (nb + 16 <= N) {
#pragma unroll
        for (int r = 0; r < 8; ++r)
            out[(size_t)(nb + r + 8 * half) * CP + l] = acc[r] + bias;
    } else {
#pragma unroll
        for (int r = 0; r < 8; ++r) {
            const int row = nb + r + 8 * half;
            if (row < N) out[(size_t)row * CP + l] = acc[r] + bias;
        }
    }
}

// ---------------- aggregation: self-loop init + edge scatter ----------------

__global__ void selfloop_init_kernel(const float* __restrict__ t, const float* __restrict__ dinv,
                                     float* __restrict__ g, long total, int logF) {
    long i = (long)blockIdx.x * blockDim.x + threadIdx.x;
    if (i >= total) return;
    const int n = (int)(i >> logF);
    const float d = dinv[n];
    g[i] = t[i] * d * d;                      // g = t * dinv^2 (self loop) == zero-init + self edge
}

// one wave per edge; edge metadata made wave-uniform (scalar loads / SALU address math)
__global__ void agg128_kernel(const float* __restrict__ t, const int* __restrict__ src,
                              const int* __restrict__ dst, const float* __restrict__ norm,
                              float* __restrict__ g, int E) {
    const int e = __builtin_amdgcn_readfirstlane(blockIdx.x * 8 + (threadIdx.x >> 5));
    if (e >= E) return;                       // uniform per wave
    const int lane = threadIdx.x & 31;
    const int r = src[e];
    const int c = dst[e];
    const float nv = norm[e];
    const float4 v = *(const float4*)(t + (size_t)r * HID + lane * 4);  // 512B/row coalesced
    float* gp = g + (size_t)c * HID + lane * 4;
    atomicAdd(gp + 0, v.x * nv);
    atomicAdd(gp + 1, v.y * nv);
    atomicAdd(gp + 2, v.z * nv);
    atomicAdd(gp + 3, v.w * nv);
}

__global__ void agg16_kernel(const float* __restrict__ t, const int* __restrict__ src,
                             const int* __restrict__ dst, const float* __restrict__ norm,
                             float* __restrict__ g, int E) {
    long tid = (long)blockIdx.x * blockDim.x + threadIdx.x;
    const int e = (int)(tid >> 2);            // 4 float4 per 16-wide row
    if (e >= E) return;
    const int fq = ((int)tid & 3) * 4;
    const int r = src[e], c = dst[e];
    const float nv = norm[e];
    const float4 v = *(const float4*)(t + (size_t)r * CP + fq);
    float* gp = g + (size_t)c * CP + fq;
    atomicAdd(gp + 0, v.x * nv);
    atomicAdd(gp + 1, v.y * nv);
    atomicAdd(gp + 2, v.z * nv);
    atomicAdd(gp + 3, v.w * nv);
}

// ---------------- sigmoid + hierarchy constraint ----------------

__global__ void constrain_kernel(const float* __restrict__ g, const float* __restrict__ R,
                                 float* __restrict__ out, int N, int C) {
    const int n = blockIdx.x * blockDim.x + threadIdx.x;
    if (n >= N) return;
    float s[CP];
#pragma unroll
    for (int j = 0; j < CP; ++j) s[j] = 0.0f;
    for (int j = 0; j < C; ++j) {
        const float v = g[(size_t)n * CP + j];
        s[j] = 1.0f / (1.0f + expf(-v));
    }
    for (int i = 0; i < C; ++i) {
        float m = -INFINITY;
        for (int j = 0; j < C; ++j) m = fmaxf(m, R[i * C + j] * s[j]);
        out[(size_t)n * C + i] = m;
    }
}

// ---------------- launch ----------------

extern "C" void kernel_launch(void* const* d_in, const int* in_sizes, int n_in,
                              void* d_out, int out_size, void* d_ws, size_t ws_size,
                              hipStream_t stream) {
    const float* x    = (const float*)d_in[0];
    const int*   eidx = (const int*)  d_in[1];   // [2,E] row-major
    const float* R    = (const float*)d_in[2];
    const float* W1   = (const float*)d_in[3];
    const float* b1   = (const float*)d_in[4];
    const float* W2   = (const float*)d_in[5];
    const float* b2   = (const float*)d_in[6];
    const float* W3   = (const float*)d_in[7];
    const float* b3   = (const float*)d_in[8];

    const int N = in_sizes[0] / DIN;
    const int E = in_sizes[1] / 2;
    const int C = out_size / N;                  // 13

    const int* src = eidx;
    const int* dst = eidx + E;

    // workspace carve-up (256B aligned)
    char* w = (char*)d_ws;
    size_t off = 0;
    auto carve = [&](size_t bytes) -> void* {
        void* p = w + off;
        off = (off + bytes + 255) & ~(size_t)255;
        return p;
    };
    float* dinv = (float*)carve((size_t)N * 4);          // aliased as uint counts first
    float* norm = (float*)carve((size_t)E * 4);
    float* bufA = (float*)carve((size_t)N * HID * 4);    // pre-aggregation features
    float* bufB = (float*)carve((size_t)N * HID * 4);    // post-aggregation features
    float* bufC = (float*)carve((size_t)N * CP * 4);     // layer-3 pre-agg (padded 16)
    float* bufD = (float*)carve((size_t)N * CP * 4);     // layer-3 post-agg
    (void)ws_size; (void)n_in;

    const int TB = 256;
    const int nTiles = (N + 15) / 16;

    // 1) degree -> dinv -> per-edge norm (computed once, reused by all 3 layers)
    hipMemsetAsync(dinv, 0, (size_t)N * 4, stream);
    count_src_kernel<<<(E + TB - 1) / TB, TB, 0, stream>>>(src, (unsigned*)dinv, E);
    make_dinv_kernel<<<(N + TB - 1) / TB, TB, 0, stream>>>(dinv, N);
    make_norm_kernel<<<(E + TB - 1) / TB, TB, 0, stream>>>(src, dst, dinv, norm, E);

    // 2) layer 1
    gemm1_wmma<<<nTiles, TB, 0, stream>>>(x, W1, b1, bufA, N);
    {
        long total = (long)N * HID;
        selfloop_init_kernel<<<(unsigned)((total + TB - 1) / TB), TB, 0, stream>>>(bufA, dinv, bufB, total, 7);
        agg128_kernel<<<(E + 7) / 8, TB, 0, stream>>>(bufA, src, dst, norm, bufB, E);
    }

    // 3) layer 2 (ReLU fused into GEMM2 LDS read)
    gemm2_wmma<<<nTiles, TB, 0, stream>>>(bufB, W2, b2, bufA, N);
    {
        long total = (long)N * HID;
        selfloop_init_kernel<<<(unsigned)((total + TB - 1) / TB), TB, 0, stream>>>(bufA, dinv, bufB, total, 7);
        agg128_kernel<<<(E + 7) / 8, TB, 0, stream>>>(bufA, src, dst, norm, bufB, E);
    }

    // 4) layer 3 (ReLU fused into GEMM3 load), padded to 16 cols
    gemm3_wmma<<<(nTiles + 7) / 8, TB, 0, stream>>>(bufB, W3, b3, bufC, N, C);
    {
        long total = (long)N * CP;
        selfloop_init_kernel<<<(unsigned)((total + TB - 1) / TB), TB, 0, stream>>>(bufC, dinv, bufD, total, 4);
        long work = (long)E * 4;
        agg16_kernel<<<(unsigned)((work + TB - 1) / TB), TB, 0, stream>>>(bufC, src, dst, norm, bufD, E);
    }

    // 5) sigmoid + constraint
    constrain_kernel<<<(N + TB - 1) / TB, TB, 0, stream>>>(bufD, R, (float*)d_out, N, C);
}